// spatial_edge_enhanced_attention_48722109006177
// MI455X (gfx1250) — compile-verified
//
#include <hip/hip_runtime.h>

typedef __attribute__((ext_vector_type(2))) float v2f;
typedef __attribute__((ext_vector_type(8))) float v8f;

#define BATCH 256
#define NJ    25
#define CIN   256
#define HH    128           // H//2
#define ROWS  (BATCH * NJ)  // 6400

// -------- Stage 1: Y[r,h] = sum_c X[r,c] * W1[h,c]  via V_WMMA_F32_16X16X4_F32
// Block = 256 threads = 8 waves. blockIdx.x selects a 16-row stripe of X;
// wave w computes the 16x16 tile at columns [16w, 16w+16).
__global__ __launch_bounds__(256)
void seea_gemm_y_wmma(const float* __restrict__ X,
                      const float* __restrict__ W1,
                      float* __restrict__ Y) {
    const int lane = threadIdx.x & 31;
    const int wave = threadIdx.x >> 5;     // 0..7 -> column tile
    const int half = lane >> 4;            // 0 or 1 (K-split across lane halves)
    const int mr   = lane & 15;            // row (A) / col (B) within tile

    const int row0 = blockIdx.x * 16;      // [0, 6400)
    const int col0 = wave * 16;            // [0, 128)

    const float* __restrict__ xrow = X  + (size_t)(row0 + mr) * CIN;
    const float* __restrict__ wrow = W1 + (size_t)(col0 + mr) * CIN;  // B[k][n] = W1[n][k]

    v8f acc = {};
#pragma unroll 8
    for (int k = 0; k < CIN; k += 4) {
        // A fragment (16x4 f32): lane(mr,half), VGPR v -> K = k + 2*half + v
        v2f a = *(const v2f*)(xrow + k + 2 * half);
        // B fragment (4x16 f32): lane n=mr, VGPR v -> K = k + 2*half + v
        v2f b = *(const v2f*)(wrow + k + 2 * half);
        acc = __builtin_amdgcn_wmma_f32_16x16x4_f32(
            /*neg_a=*/false, a, /*neg_b=*/false, b,
            /*c_mod=*/(short)0, acc, /*reuse_a=*/false, /*reuse_b=*/false);
    }

    // D layout: lane n = mr, VGPR v -> m = v + 8*half
#pragma unroll
    for (int v = 0; v < 8; ++v) {
        const int m = v + 8 * half;
        Y[(size_t)(row0 + m) * HH + col0 + mr] = acc[v];
    }
}

// -------- Stage 2: out[b,i,j] = sum_h prelu(Y[b,j,h]-Y[b,i,h]) * W2[h]
// One block per batch element; Y[b] (25*128 f32 = 12.8 KB) + W2 staged in LDS.
__global__ __launch_bounds__(256)
void seea_pair_prelu_dot(const float* __restrict__ Y,
                         const float* __restrict__ alpha_p,
                         const float* __restrict__ W2,
                         float* __restrict__ out) {
    __shared__ float ys[NJ * HH];
    __shared__ float w2s[HH];

    const int b   = blockIdx.x;
    const int tid = threadIdx.x;
    const float alpha = alpha_p[0];

    const float* __restrict__ yb = Y + (size_t)b * NJ * HH;
    for (int t = tid; t < NJ * HH; t += 256) ys[t] = yb[t];
    if (tid < HH) w2s[tid] = W2[tid];
    __syncthreads();

    for (int p = tid; p < NJ * NJ; p += 256) {
        const int i = p / NJ;
        const int j = p - i * NJ;
        const float* __restrict__ yi = ys + i * HH;
        const float* __restrict__ yj = ys + j * HH;
        float s = 0.f;
#pragma unroll 8
        for (int h = 0; h < HH; ++h) {
            float d = yj[h] - yi[h];
            d = (d >= 0.f) ? d : alpha * d;
            s = fmaf(d, w2s[h], s);
        }
        out[(size_t)b * (NJ * NJ) + p] = s;
    }
}

extern "C" void kernel_launch(void* const* d_in, const int* in_sizes, int n_in,
                              void* d_out, int out_size, void* d_ws, size_t ws_size,
                              hipStream_t stream) {
    // setup_inputs order: src, heads, ends, pair_ids, W1, alpha, W2
    const float* src   = (const float*)d_in[0];
    const float* W1    = (const float*)d_in[4];   // [128, 256]
    const float* alpha = (const float*)d_in[5];   // [1]
    const float* W2    = (const float*)d_in[6];   // [1, 128] -> 128 floats
    float* Y   = (float*)d_ws;                    // [6400, 128] scratch (3.27 MB)
    float* out = (float*)d_out;                   // [256, 25, 25, 1]

    (void)in_sizes; (void)n_in; (void)out_size; (void)ws_size;

    seea_gemm_y_wmma<<<ROWS / 16, 256, 0, stream>>>(src, W1, Y);
    seea_pair_prelu_dot<<<BATCH, 256, 0, stream>>>(Y, alpha, W2, out);
}